// OrthogonalButterfly_85220741087978
// MI455X (gfx1250) — compile-verified
//
#include <hip/hip_runtime.h>
#include <stdint.h>

// Orthogonal butterfly network, 10 layers over 32768x1024 f32.
// Memory-bound: 268 MB of HBM traffic -> ~11.5 us floor at 23.3 TB/s.
// GEMM/WMMA reformulation would cost 68x the FLOPs; butterfly + TDM tile
// loads + LDS ping-pong is the roofline-optimal CDNA5 path.
//
// NOTE: main kernel is defined FIRST so the disasm snippet shows its
// prologue (TDM descriptor build + tensor_load_to_lds + s_wait_tensorcnt).

#define WIDTH   1024
#define HALF    512
#define DEPTH   10
#define BATCH   32768
#define ROWS    8
#define THREADS 256

typedef __attribute__((ext_vector_type(4))) unsigned int v4u;
typedef __attribute__((ext_vector_type(8))) int          v8i;
typedef __attribute__((ext_vector_type(4))) int          v4i;

// ---------------------------------------------------------------------------
// Main butterfly kernel. One block = 8 rows. TDM async tile load into LDS,
// 9 ping-pong layers in LDS, layer 10 fused with bias add -> coalesced
// float2 global stores.
// ---------------------------------------------------------------------------
__global__ void __launch_bounds__(THREADS)
butterfly_main_kernel(const float* __restrict__ X,
                      const float* __restrict__ csc,
                      const float* __restrict__ css,
                      const float* __restrict__ bias,
                      float* __restrict__ out) {
    __shared__ float lds[2][ROWS * WIDTH];   // 64 KB -> 4 blocks per WGP

    const int t = threadIdx.x;
    const long long r0 = (long long)blockIdx.x * ROWS;

    // ---- TDM: async load 8x1024 f32 tile (32 KB) into lds[0] --------------
    if (t < 32) {  // wave 0 issues; TDM ignores EXEC, issues once per wave
        uint32_t lds_base = (uint32_t)(uintptr_t)(&lds[0][0]);  // low 32b of
                                                                // generic ptr
                                                                // = LDS offset
        uint64_t ga = (uint64_t)(uintptr_t)(X + r0 * WIDTH);

        v4u g0;
        g0[0] = 1u;                                   // count=1, user mode
        g0[1] = lds_base;                             // lds_addr
        g0[2] = (uint32_t)(ga & 0xFFFFFFFFu);         // global_addr[31:0]
        g0[3] = (uint32_t)((ga >> 32) & 0x1FFFFFFu)   // global_addr[56:32]
              | (2u << 30);                           // type = 2 ("image")

        v8i g1;
        g1[0] = (int)(2u << 16);                      // data_size = 4 bytes
        g1[1] = (int)((WIDTH & 0xFFFF) << 16);        // tensor_dim0[15:0]
        g1[2] = (int)((WIDTH >> 16) | (ROWS << 16));  // t_dim0[31:16], t_dim1[15:0]
        g1[3] = (int)((WIDTH & 0xFFFF) << 16);        // t_dim1[31:16]=0, tile_dim0=1024
        g1[4] = (int)ROWS;                            // tile_dim1=8, tile_dim2=0
        g1[5] = (int)WIDTH;                           // tensor_dim0_stride[31:0]
        g1[6] = 0;                                    // stride hi, dim1_stride lo
        g1[7] = 0;

        v4i gz4 = {0, 0, 0, 0};                       // groups 2/3 unused (2-D)
        v8i gz8 = {0, 0, 0, 0, 0, 0, 0, 0};           // extra group (clang-23 form)

        __builtin_amdgcn_tensor_load_to_lds(g0, g1, gz4, gz4, gz8, 0);
        __builtin_amdgcn_s_wait_tensorcnt(0);
    }
    __syncthreads();

    // ---- layers 0..8: LDS ping-pong --------------------------------------
    // Each thread handles pairs i = t and i = t+256 for all 8 rows.
    // Layer d reads lds[d&1], writes lds[1-(d&1)].
#pragma unroll
    for (int d = 0; d < DEPTH - 1; ++d) {
        const float c0 = csc[d * HALF + t];
        const float s0 = css[d * HALF + t];
        const float c1 = csc[d * HALF + t + 256];
        const float s1 = css[d * HALF + t + 256];
        const float* in  = (d & 1) ? lds[1] : lds[0];
        float*       ob  = (d & 1) ? lds[0] : lds[1];
#pragma unroll
        for (int r = 0; r < ROWS; ++r) {
            const float* rowin = in + r * WIDTH;
            float2* rowout = (float2*)(ob + r * WIDTH);
            float x0a = rowin[t];
            float x1a = rowin[t + HALF];
            float x0b = rowin[t + 256];
            float x1b = rowin[t + 256 + HALF];
            // new0 = x0*c + x1*s ; new1 = x1*c - x0*s ; interleave at 2i,2i+1
            rowout[t]       = make_float2(fmaf(x0a, c0,  x1a * s0),
                                          fmaf(x1a, c0, -x0a * s0));
            rowout[t + 256] = make_float2(fmaf(x0b, c1,  x1b * s1),
                                          fmaf(x1b, c1, -x0b * s1));
        }
        __syncthreads();
    }

    // ---- layer 9: fused with bias add, direct coalesced global stores ----
    {
        const int d = DEPTH - 1;
        const float c0 = csc[d * HALF + t];
        const float s0 = css[d * HALF + t];
        const float c1 = csc[d * HALF + t + 256];
        const float s1 = css[d * HALF + t + 256];
        const float2 b0 = ((const float2*)bias)[t];        // bias[2t], bias[2t+1]
        const float2 b1 = ((const float2*)bias)[t + 256];
        const float* in = lds[1];                          // layer 8 wrote lds[1]
#pragma unroll
        for (int r = 0; r < ROWS; ++r) {
            const float* rowin = in + r * WIDTH;
            float2* rowout = (float2*)(out + (r0 + r) * WIDTH);
            float x0a = rowin[t];
            float x1a = rowin[t + HALF];
            float x0b = rowin[t + 256];
            float x1b = rowin[t + 256 + HALF];
            rowout[t]       = make_float2(fmaf(x0a, c0,  x1a * s0) + b0.x,
                                          fmaf(x1a, c0, -x0a * s0) + b0.y);
            rowout[t + 256] = make_float2(fmaf(x0b, c1,  x1b * s1) + b1.x,
                                          fmaf(x1b, c1, -x0b * s1) + b1.y);
        }
    }
}

// ---------------------------------------------------------------------------
// Precompute cos/sin tables, laid out [d][i] for unit-stride reads.
// params is (HALF, DEPTH) row-major: params[i*DEPTH + d].
// ---------------------------------------------------------------------------
__global__ void butterfly_cs_kernel(const float* __restrict__ params,
                                    float* __restrict__ csc,
                                    float* __restrict__ css) {
    int j = blockIdx.x * blockDim.x + threadIdx.x;
    if (j < HALF * DEPTH) {
        int i = j / DEPTH;
        int d = j - i * DEPTH;
        float v = params[j];
        csc[d * HALF + i] = cosf(v);
        css[d * HALF + i] = sinf(v);
    }
}

// ---------------------------------------------------------------------------
extern "C" void kernel_launch(void* const* d_in, const int* in_sizes, int n_in,
                              void* d_out, int out_size, void* d_ws, size_t ws_size,
                              hipStream_t stream) {
    const float* X      = (const float*)d_in[0];   // (32768, 1024) f32
    const float* params = (const float*)d_in[1];   // (512, 10) f32
    const float* bias   = (const float*)d_in[2];   // (1024,) f32
    float* out = (float*)d_out;                    // (32768, 1024) f32

    float* csc = (float*)d_ws;                     // [DEPTH][HALF]
    float* css = csc + HALF * DEPTH;               // [DEPTH][HALF]  (40 KB total)

    butterfly_cs_kernel<<<(HALF * DEPTH + 255) / 256, 256, 0, stream>>>(
        params, csc, css);

    butterfly_main_kernel<<<BATCH / ROWS, THREADS, 0, stream>>>(
        X, csc, css, bias, out);
}